// TransR_49881750176020
// MI455X (gfx1250) — compile-verified
//
#include <hip/hip_runtime.h>

// TransR margin loss for MI455X (gfx1250, wave32, WMMA).
//
// Per triple i: out(4x128) = [h_pos; t_pos; h_neg; t_neg] (4x128) @ P[r] (128x128)
// via V_WMMA_F32_16X16X4_F32 (M=16 tile, rows 4..15 zero; 8 N-tiles; 32 K-steps).
// The A-staging LDS buffer holds all 16 M-rows with rows 4..15 zero-filled so
// the inner-loop A fetch is a branchless ds_load_b64 (no EXEC save/restore).
// Then dist = ||h_proj + r_emb - t_proj||, loss_i = relu(1 + d_pos - d_neg),
// summed deterministically by a second kernel.

typedef __attribute__((ext_vector_type(2))) float v2f;
typedef __attribute__((ext_vector_type(8))) float v8f;

#define TR_DIM 128
#define TR_MARGIN 1.0f
#define LDS_STRIDE 132   // 132 % 64 == 4 -> the 16 A-rows land on distinct banks

__launch_bounds__(32)
__global__ void transr_wmma_kernel(const int* __restrict__ pos,
                                   const int* __restrict__ neg,
                                   const float* __restrict__ ent,
                                   const float* __restrict__ rel,
                                   const float* __restrict__ proj,
                                   float* __restrict__ loss,
                                   int batch)
{
    const int i = blockIdx.x;
    if (i >= batch) return;

    const int lane = threadIdx.x;        // 0..31 (wave32)

    // Uniform (scalar) gathers of the 5 indices for this triple.
    const int hp = pos[i * 3 + 0];
    const int tp = pos[i * 3 + 1];
    const int r  = pos[i * 3 + 2];       // r_neg == r_pos by construction
    const int hn = neg[i * 3 + 0];
    const int tn = neg[i * 3 + 1];

    // Stage the A matrix in LDS: rows 0..3 = {hp, tp, hn, tn} embeddings,
    // rows 4..15 zero (so every lane's A fetch is unconditional).
    __shared__ float ldsV[16 * LDS_STRIDE];
    {
        const float* __restrict__ vhp = ent + (size_t)hp * TR_DIM;
        const float* __restrict__ vtp = ent + (size_t)tp * TR_DIM;
        const float* __restrict__ vhn = ent + (size_t)hn * TR_DIM;
        const float* __restrict__ vtn = ent + (size_t)tn * TR_DIM;
        for (int d = lane; d < TR_DIM; d += 32) {
            ldsV[0 * LDS_STRIDE + d] = vhp[d];
            ldsV[1 * LDS_STRIDE + d] = vtp[d];
            ldsV[2 * LDS_STRIDE + d] = vhn[d];
            ldsV[3 * LDS_STRIDE + d] = vtn[d];
            #pragma unroll
            for (int v = 4; v < 16; ++v) {
                ldsV[v * LDS_STRIDE + d] = 0.f;
            }
        }
    }
    __syncthreads();

    // 8 accumulator tiles cover all 128 output dims: D(16x16) per tile,
    // rows M=0..3 = {hp,tp,hn,tn} projections, M=4..15 stay zero.
    v8f acc[8];
    #pragma unroll
    for (int e = 0; e < 8; ++e) acc[e] = (v8f){0.f,0.f,0.f,0.f,0.f,0.f,0.f,0.f};

    // ISA f32 16x16x4 operand layout (A 16x4, B 4x16, both symmetric):
    //   element (j, k): lane = j + 16*(k>=2), vgpr = k&1.
    const int m  = lane & 15;            // A row (M) for this lane
    const int n  = lane & 15;            // B col (N) for this lane
    const int kh = lane >> 4;            // 0 -> K={0,1}, 1 -> K={2,3}

    const float* __restrict__ Pr  = proj + (size_t)r * (TR_DIM * TR_DIM);
    const float* __restrict__ Av  = &ldsV[m * LDS_STRIDE + 2 * kh];

    for (int kk = 0; kk < 32; ++kk) {    // K = 128 in steps of 4
        const int d0 = kk * 4;

        // A operand: ldsV[m][d0 + 2*kh + {0,1}] — branchless ds_load_b64.
        v2f a;
        a.x = Av[d0];
        a.y = Av[d0 + 1];

        // B operand base: P[d0 + 2*kh][n], row stride TR_DIM.
        const float* __restrict__ Pb = Pr + (size_t)(d0 + 2 * kh) * TR_DIM + n;

        #pragma unroll
        for (int e = 0; e < 8; ++e) {
            v2f b;
            b.x = Pb[e * 16];            // P[d0+2*kh  ][e*16 + n]
            b.y = Pb[TR_DIM + e * 16];   // P[d0+2*kh+1][e*16 + n]
            acc[e] = __builtin_amdgcn_wmma_f32_16x16x4_f32(
                /*neg_a=*/false, a, /*neg_b=*/false, b,
                /*c_mod=*/(short)0, acc[e],
                /*reuse_a=*/false, /*reuse_b=*/false);
        }
    }

    // C/D layout: D[m][n] -> vgpr m (m<8), lane n (lanes 0..15).
    // So lanes 0..15 hold out[v][e*16 + lane] in acc[e][v], v in 0..3.
    float sp = 0.f, sn = 0.f;
    if (lane < 16) {
        const float* __restrict__ re = rel + (size_t)r * TR_DIM + lane;
        #pragma unroll
        for (int e = 0; e < 8; ++e) {
            const float rv = re[e * 16];
            const float dp = acc[e][0] + rv - acc[e][1];   // h_pos_proj + r - t_pos_proj
            const float dn = acc[e][2] + rv - acc[e][3];   // h_neg_proj + r - t_neg_proj
            sp += dp * dp;
            sn += dn * dn;
        }
    }

    // Wave reduction (upper 16 lanes contribute zero).
    #pragma unroll
    for (int off = 16; off >= 1; off >>= 1) {
        sp += __shfl_xor(sp, off, 32);
        sn += __shfl_xor(sn, off, 32);
    }

    if (lane == 0) {
        loss[i] = fmaxf(0.f, TR_MARGIN + sqrtf(sp) - sqrtf(sn));
    }
}

// Deterministic single-block sum of the per-triple losses.
__launch_bounds__(256)
__global__ void transr_reduce_kernel(const float* __restrict__ in, int n,
                                     float* __restrict__ out)
{
    __shared__ float smem[256];
    float v = 0.f;
    for (int i = threadIdx.x; i < n; i += 256) v += in[i];
    smem[threadIdx.x] = v;
    __syncthreads();
    #pragma unroll
    for (int off = 128; off > 0; off >>= 1) {
        if ((int)threadIdx.x < off) smem[threadIdx.x] += smem[threadIdx.x + off];
        __syncthreads();
    }
    if (threadIdx.x == 0) out[0] = smem[0];
}

extern "C" void kernel_launch(void* const* d_in, const int* in_sizes, int n_in,
                              void* d_out, int out_size, void* d_ws, size_t ws_size,
                              hipStream_t stream)
{
    // setup_inputs() order: positive_triples, negative_triples,
    //                       entity_emb, relation_emb, proj_matrix
    const int*   pos  = (const int*)d_in[0];
    const int*   neg  = (const int*)d_in[1];
    const float* ent  = (const float*)d_in[2];
    const float* rel  = (const float*)d_in[3];
    const float* proj = (const float*)d_in[4];
    float*       out  = (float*)d_out;
    float*       ws   = (float*)d_ws;           // batch floats of scratch

    const int batch = in_sizes[0] / 3;          // 16384

    transr_wmma_kernel<<<batch, 32, 0, stream>>>(pos, neg, ent, rel, proj, ws, batch);
    transr_reduce_kernel<<<1, 256, 0, stream>>>(ws, batch, out);
}